// KG_GNN_84430467105347
// MI455X (gfx1250) — compile-verified
//
#include <hip/hip_runtime.h>

#define N_NODES 100000
#define IN_DIM  128
#define HEADS   4
#define HC      128   // HEADS * OUT_C
#define NEG_SLOPE 0.2f

typedef float v2f __attribute__((ext_vector_type(2)));
typedef float v8f __attribute__((ext_vector_type(8)));

// monotone float<->int map so atomicMax(int) implements float max (self-inverse)
__device__ __forceinline__ int f2o(float f) {
    int i = __float_as_int(f);
    return i >= 0 ? i : (i ^ 0x7FFFFFFF);
}
__device__ __forceinline__ float o2f(int i) {
    int j = i >= 0 ? i : (i ^ 0x7FFFFFFF);
    return __int_as_float(j);
}

// ---------------------------------------------------------------------------
// Dual GEMM via fp32 WMMA: x_l = x@W_l + b_l, x_r = x@W_r + b_r.
// One wave -> 16 rows x 128 cols for ONE matrix: 8 accumulator tiles share a
// single A load per k-step (8 independent v_wmma back-to-back, 8x A reuse).
// Tasks = 6250 rowTiles * 2 matrices = 12500 waves; 4 waves/block -> 3125 blocks.
// ---------------------------------------------------------------------------
__global__ void gat_gemm(const float* __restrict__ x,
                         const float* __restrict__ Wl, const float* __restrict__ bl,
                         const float* __restrict__ Wr, const float* __restrict__ br,
                         float* __restrict__ xl, float* __restrict__ xr)
{
    const int wid  = threadIdx.x >> 5;          // 0..3
    const int lane = threadIdx.x & 31;
    const int task = blockIdx.x * 4 + wid;      // 12500 tasks
    const int matSel  = task & 1;               // 0 = W_l, 1 = W_r
    const int rowTile = task >> 1;              // 0..6249

    const float* W    = matSel ? Wr : Wl;
    const float* bv   = matSel ? br : bl;
    float*       outp = matSel ? xr : xl;

    const int row0 = rowTile * 16;
    const int lm = lane & 15;          // M index (A) / N index (B)
    const int lk = (lane >> 4) * 2;    // K sub-offset per ISA VGPR layout

    const float* aptr = x + (size_t)(row0 + lm) * IN_DIM + lk;
    const float* bptr = W + (size_t)lk * HC + lm;

    v8f c[8] = {};
    for (int k0 = 0; k0 < IN_DIM; k0 += 4) {
        const float2 av = *(const float2*)aptr;   // A[m][k0+lk], A[m][k0+lk+1]
        v2f a; a.x = av.x; a.y = av.y;
        #pragma unroll
        for (int t = 0; t < 8; ++t) {             // 8 column tiles of 16
            v2f b;
            b.x = bptr[t * 16];                   // B[k0+lk  ][16t+n]
            b.y = bptr[HC + t * 16];              // B[k0+lk+1][16t+n]
            c[t] = __builtin_amdgcn_wmma_f32_16x16x4_f32(
                       false, a, false, b, (short)0, c[t], false, false);
        }
        aptr += 4;
        bptr += 4 * HC;
    }

    const int rbase = row0 + ((lane >> 4) * 8);   // lanes 16-31 hold M=8..15
    #pragma unroll
    for (int t = 0; t < 8; ++t) {
        const int col = t * 16 + lm;
        const float bb = bv[col];
        #pragma unroll
        for (int r = 0; r < 8; ++r)
            outp[(size_t)(rbase + r) * HC + col] = c[t][r] + bb;
    }
}

// ---------------------------------------------------------------------------
// Init: zero numerator accumulator (d_out) and denom; segmax = INT_MIN.
// ---------------------------------------------------------------------------
__global__ void gat_init(float* __restrict__ out, float* __restrict__ denom,
                         int* __restrict__ segmax, int nOut, int nSeg)
{
    int i = blockIdx.x * blockDim.x + threadIdx.x;
    if (i < nOut) out[i] = 0.0f;
    if (i < nSeg) { denom[i] = 0.0f; segmax[i] = (int)0x80000000; }
}

// ---------------------------------------------------------------------------
// Edge kernels: 32 lanes per edge. lane owns channels [4*lane .. 4*lane+3],
// head = lane>>3. Per-head dot with att reduced over 8 lanes via shfl_xor.
// x_l + x_r = 102 MB -> resident in the 192 MB L2, so gathers are L2 hits.
// ---------------------------------------------------------------------------
__device__ __forceinline__ float edge_score(const float4 a, const float4 b,
                                            const float4 at)
{
    float m0 = a.x + b.x; m0 = m0 >= 0.0f ? m0 : NEG_SLOPE * m0;
    float m1 = a.y + b.y; m1 = m1 >= 0.0f ? m1 : NEG_SLOPE * m1;
    float m2 = a.z + b.z; m2 = m2 >= 0.0f ? m2 : NEG_SLOPE * m2;
    float m3 = a.w + b.w; m3 = m3 >= 0.0f ? m3 : NEG_SLOPE * m3;
    float p = at.x * m0 + at.y * m1 + at.z * m2 + at.w * m3;
    p += __shfl_xor(p, 1, 32);
    p += __shfl_xor(p, 2, 32);
    p += __shfl_xor(p, 4, 32);
    return p;   // full per-head sum present in all 8 lanes of the group
}

__global__ void gat_edge_max(const float* __restrict__ xl, const float* __restrict__ xr,
                             const float* __restrict__ att,
                             const int* __restrict__ src, const int* __restrict__ dst,
                             int nEdges, int nTot, int* __restrict__ segmax)
{
    const int e = blockIdx.x * 8 + (threadIdx.x >> 5);
    if (e >= nTot) return;
    const int lane = threadIdx.x & 31;
    int s, d;
    if (e < nEdges) { s = src[e]; d = dst[e]; } else { s = e - nEdges; d = s; }

    const float4 a  = *(const float4*)(xl + (size_t)s * HC + lane * 4);
    const float4 b  = *(const float4*)(xr + (size_t)d * HC + lane * 4);
    const float4 at = *(const float4*)(att + lane * 4);
    const float p = edge_score(a, b, at);

    if ((lane & 7) == 0)
        atomicMax(segmax + (size_t)d * HEADS + (lane >> 3), f2o(p));
}

__global__ void gat_edge_acc(const float* __restrict__ xl, const float* __restrict__ xr,
                             const float* __restrict__ att,
                             const int* __restrict__ src, const int* __restrict__ dst,
                             int nEdges, int nTot,
                             const int* __restrict__ segmax,
                             float* __restrict__ denom, float* __restrict__ out)
{
    const int e = blockIdx.x * 8 + (threadIdx.x >> 5);
    if (e >= nTot) return;
    const int lane = threadIdx.x & 31;
    int s, d;
    if (e < nEdges) { s = src[e]; d = dst[e]; } else { s = e - nEdges; d = s; }

    const float4 a  = *(const float4*)(xl + (size_t)s * HC + lane * 4);
    const float4 b  = *(const float4*)(xr + (size_t)d * HC + lane * 4);
    const float4 at = *(const float4*)(att + lane * 4);
    const float p = edge_score(a, b, at);

    const int h = lane >> 3;
    const float mx = o2f(segmax[(size_t)d * HEADS + h]);
    const float w = __expf(p - mx);

    if ((lane & 7) == 0)
        unsafeAtomicAdd(denom + (size_t)d * HEADS + h, w);

    float* o = out + (size_t)d * HC + lane * 4;
    unsafeAtomicAdd(o + 0, w * a.x);
    unsafeAtomicAdd(o + 1, w * a.y);
    unsafeAtomicAdd(o + 2, w * a.z);
    unsafeAtomicAdd(o + 3, w * a.w);
}

// ---------------------------------------------------------------------------
// Finalize: out = numerator / denom + bias
// ---------------------------------------------------------------------------
__global__ void gat_final(float* __restrict__ out, const float* __restrict__ denom,
                          const float* __restrict__ bias, int nOut)
{
    int i = blockIdx.x * blockDim.x + threadIdx.x;
    if (i >= nOut) return;
    const int node = i >> 7;
    const int ch   = i & 127;
    out[i] = out[i] / denom[node * HEADS + (ch >> 5)] + bias[ch];
}

// ---------------------------------------------------------------------------
extern "C" void kernel_launch(void* const* d_in, const int* in_sizes, int n_in,
                              void* d_out, int out_size, void* d_ws, size_t ws_size,
                              hipStream_t stream)
{
    const float* x    = (const float*)d_in[0];
    const float* Wl   = (const float*)d_in[1];
    const float* bl   = (const float*)d_in[2];
    const float* Wr   = (const float*)d_in[3];
    const float* br   = (const float*)d_in[4];
    const float* att  = (const float*)d_in[5];
    const float* bias = (const float*)d_in[6];
    const int*   src  = (const int*)d_in[7];
    const int*   dst  = (const int*)d_in[8];
    const int nE = in_sizes[7];
    float* out = (float*)d_out;

    // workspace layout: x_l | x_r | denom | segmax  (~106 MB)
    float* xl    = (float*)d_ws;
    float* xr    = xl + (size_t)N_NODES * HC;
    float* denom = xr + (size_t)N_NODES * HC;
    int*   smax  = (int*)(denom + (size_t)N_NODES * HEADS);

    const int nOut = N_NODES * HC;
    const int nSeg = N_NODES * HEADS;

    gat_init<<<(nOut + 255) / 256, 256, 0, stream>>>(out, denom, smax, nOut, nSeg);

    // 12500 wave-tasks, 4 waves per 128-thread block -> 3125 blocks
    gat_gemm<<<3125, 128, 0, stream>>>(x, Wl, bl, Wr, br, xl, xr);

    const int nTot = nE + N_NODES;
    const int eb = (nTot + 7) / 8;
    gat_edge_max<<<eb, 256, 0, stream>>>(xl, xr, att, src, dst, nE, nTot, smax);
    gat_edge_acc<<<eb, 256, 0, stream>>>(xl, xr, att, src, dst, nE, nTot, smax, denom, out);

    gat_final<<<(nOut + 255) / 256, 256, 0, stream>>>(out, denom, bias, nOut);
}